// CgcnNodeUpdate_24412594110749
// MI455X (gfx1250) — compile-verified
//
#include <hip/hip_runtime.h>

// Problem constants (fixed by the reference harness).
constexpr int BATCH  = 2;
constexpr int NNODES = 50000;
constexpr int CD     = 128;              // COMP_DIM
constexpr int OD     = 128;              // OUT_DIM
constexpr int SEGS   = BATCH * NNODES;   // 100000 segments

typedef __attribute__((ext_vector_type(2))) float v2f;
typedef __attribute__((ext_vector_type(8))) float v8f;

// ---------------------------------------------------------------------------
// Kernel 0: zero the accumulator (lives in d_out) and the counts (in d_ws).
// ---------------------------------------------------------------------------
__global__ void zero_kernel(float* __restrict__ acc, float* __restrict__ counts) {
    size_t stride = (size_t)gridDim.x * blockDim.x;
    size_t i      = (size_t)blockIdx.x * blockDim.x + threadIdx.x;
    size_t total  = (size_t)SEGS * OD;
    for (size_t j = i; j < total; j += stride) acc[j] = 0.0f;
    for (size_t j = i; j < (size_t)SEGS; j += stride) counts[j] = 0.0f;
}

// ---------------------------------------------------------------------------
// Kernel 1: edge scatter.  One edge per 32-lane group; each lane owns a
// float4 slice of the 128-dim comp vector (h_src - r_rel) and scatter-adds
// it into acc[b*N + tgt] with native f32 atomics (L2-resident target).
// edge_indices layout: [4, E] row-major -> rows b, tgt, src, rel.
// ---------------------------------------------------------------------------
__global__ void edge_scatter_kernel(const float* __restrict__ node,
                                    const int*   __restrict__ idx,
                                    const float* __restrict__ rel,
                                    float* __restrict__ acc,
                                    float* __restrict__ counts,
                                    int E) {
    int lane = threadIdx.x & 31;
    int e    = blockIdx.x * (blockDim.x >> 5) + (threadIdx.x >> 5);
    if (e >= E) return;

    int b = idx[e];
    int t = idx[(size_t)E + e];
    int s = idx[2 * (size_t)E + e];
    int r = idx[3 * (size_t)E + e];

    const float4* hp = (const float4*)(node + ((size_t)b * NNODES + s) * CD);
    const float4* rp = (const float4*)(rel + (size_t)r * CD);
    float4 hv = hp[lane];
    float4 rv = rp[lane];

    float* dst = acc + ((size_t)b * NNODES + t) * CD + lane * 4;
    unsafeAtomicAdd(dst + 0, hv.x - rv.x);
    unsafeAtomicAdd(dst + 1, hv.y - rv.y);
    unsafeAtomicAdd(dst + 2, hv.z - rv.z);
    unsafeAtomicAdd(dst + 3, hv.w - rv.w);
    if (lane == 0) unsafeAtomicAdd(counts + (size_t)b * NNODES + t, 1.0f);
}

// ---------------------------------------------------------------------------
// Kernel 2: in-place projection + mean.  Each block owns 16 segment rows.
// Stage acc[16 x 128] into LDS (padded rows -> conflict-free), barrier, then
// 8 waves each produce a 16x16 output tile via chained
// V_WMMA_F32_16X16X4_F32 over K=128, divide by count, store back in place.
//   out[m][n] = (1/cnt[m]) * sum_k acc[m][k] * W[n][k]
// A-frag (16x4 f32): a[j] = A[lane%16][k + 2*(lane>=16) + j]
// B-frag (4x16 f32): b[j] = B[k + 2*(lane>=16) + j][lane%16] = W[n][...]
// C/D   (16x16 f32): c[v] -> M = v + 8*(lane>=16), N = lane%16
// ---------------------------------------------------------------------------
__global__ __launch_bounds__(256) void project_mean_kernel(
        float* __restrict__ acc_out,        // [SEGS, CD] in, [SEGS, OD] out
        const float* __restrict__ W,        // [OD, CD]
        const float* __restrict__ counts) { // [SEGS]
    constexpr int PAD = CD + 4;             // 132 floats: stride%64banks == 4
    __shared__ __align__(16) float accS[16][PAD];
    __shared__ float cntS[16];

    int tid     = threadIdx.x;
    int rowBase = blockIdx.x * 16;

    // Cooperative stage: 16*128 floats = 512 float4, 2 per thread.
    const float4* src = (const float4*)(acc_out + (size_t)rowBase * CD);
    #pragma unroll
    for (int i = tid; i < 16 * CD / 4; i += 256) {
        float4 v = src[i];
        int rr = i / (CD / 4);
        int c4 = i % (CD / 4);
        *((float4*)&accS[rr][c4 * 4]) = v;
    }
    if (tid < 16) cntS[tid] = counts[rowBase + tid];
    __syncthreads();

    int wave = tid >> 5;
    int lane = tid & 31;
    int hi   = lane >> 4;   // 0 or 1
    int lo   = lane & 15;
    int n0   = wave * 16;   // output-column tile for this wave

    v8f c = {};
    #pragma unroll
    for (int k = 0; k < CD; k += 4) {
        v2f a, b;
        a.x = accS[lo][k + 2 * hi + 0];
        a.y = accS[lo][k + 2 * hi + 1];
        const float* wp = W + (size_t)(n0 + lo) * CD + k + 2 * hi;
        b.x = wp[0];
        b.y = wp[1];
        c = __builtin_amdgcn_wmma_f32_16x16x4_f32(
                /*neg_a=*/false, a, /*neg_b=*/false, b,
                /*c_mod=*/(short)0, c, /*reuse_a=*/false, /*reuse_b=*/false);
    }

    #pragma unroll
    for (int v = 0; v < 8; ++v) {
        int m = v + 8 * hi;
        float o = c[v] / cntS[m];
        acc_out[(size_t)(rowBase + m) * OD + n0 + lo] = o;
    }
}

// ---------------------------------------------------------------------------
extern "C" void kernel_launch(void* const* d_in, const int* in_sizes, int n_in,
                              void* d_out, int out_size, void* d_ws, size_t ws_size,
                              hipStream_t stream) {
    const float* node  = (const float*)d_in[0];   // [B, N, CD] f32
    const int*   idx   = (const int*)d_in[1];     // [4, E] int
    const float* rel   = (const float*)d_in[2];   // [N_REL, CD] f32
    const float* W     = (const float*)d_in[3];   // [OD, CD] f32
    float*       out   = (float*)d_out;           // [B, N, OD] f32 (also acc)
    float*       cnts  = (float*)d_ws;            // [SEGS] f32 scratch

    int E = in_sizes[1] / 4;

    zero_kernel<<<2048, 256, 0, stream>>>(out, cnts);

    int edgesPerBlock = 256 / 32;
    edge_scatter_kernel<<<(E + edgesPerBlock - 1) / edgesPerBlock, 256, 0, stream>>>(
        node, idx, rel, out, cnts, E);

    project_mean_kernel<<<SEGS / 16, 256, 0, stream>>>(out, W, cnts);
}